// PermutoEncoding_84576495993515
// MI455X (gfx1250) — compile-verified
//
#include <hip/hip_runtime.h>

// Permutohedral hash-grid encoding for MI455X (gfx1250, wave32).
// Gather/latency-bound op: no matmul structure -> no WMMA (a 4x2 blend per
// thread cannot feed 16x16x32 tiles usefully). Optimization targets instead:
//  - keep the 48MB lattice table resident in the 192MB L2 (default RT loads)
//  - non-temporal (TH_NT) stores for the 96MB output stream so it does not
//    evict the table from L2
//  - full coalescing: thread t -> (point n = t/24, level l = t%24); a wave32
//    writes one contiguous 256B run and broadcasts ~2 position vectors
//  - per-level constants staged in LDS
//  - hash offsets computed early + global_prefetch_b8 on all 4 gather lines,
//    so the barycentric VALU work overlaps the L2 gather latency

typedef float v2f __attribute__((ext_vector_type(2)));

#define POS_DIM   3
#define DP1       4
#define CAPACITY  262144u     // 2^18 -> modulo is a mask
#define NR_LEVELS 24
#define NR_FEAT   2

// ---------------------------------------------------------------------------
// Init: scale_factor[l][j] = float32(10^(3l/23)) * float32(1/sqrt((j+1)(j+2)))
// Computed in double to match numpy geomspace after the f32 cast.
// Runs every launch (deterministic, graph-capture safe), 24 threads.
// ---------------------------------------------------------------------------
__global__ void permuto_init_sf(float* __restrict__ sf)
{
    int l = threadIdx.x;
    if (l >= NR_LEVELS) return;
    double lin = (3.0 * (double)l) / 23.0;
    float s = (float)pow(10.0, lin);          // SCALES[l] as f32
#pragma unroll
    for (int j = 0; j < POS_DIM; ++j) {
        float inv = (float)(1.0 / sqrt((double)((j + 1) * (j + 2))));
        sf[l * POS_DIM + j] = s * inv;        // f32 product, as in reference
    }
}

// ---------------------------------------------------------------------------
// Main kernel: one thread per (point, level).
// ---------------------------------------------------------------------------
__global__ __launch_bounds__(256) void permuto_encode(
    const float* __restrict__ positions,   // [N,3]
    const float* __restrict__ anneal,      // [24]
    const float* __restrict__ lattice,     // [24, 262144, 2]
    const float* __restrict__ shift,       // [24,3]
    const float* __restrict__ sf,          // [24,3] precomputed scale factors
    float* __restrict__ out,               // [N, 48]
    unsigned n_points)
{
    __shared__ float s_sf[NR_LEVELS * POS_DIM];
    __shared__ float s_shift[NR_LEVELS * POS_DIM];
    __shared__ float s_anneal[NR_LEVELS];

    const int tid = threadIdx.x;
    if (tid < NR_LEVELS * POS_DIM) {
        s_sf[tid]    = sf[tid];
        s_shift[tid] = shift[tid];
    }
    if (tid < NR_LEVELS) s_anneal[tid] = anneal[tid];
    __syncthreads();

    const unsigned t     = blockIdx.x * blockDim.x + tid;
    const unsigned total = n_points * NR_LEVELS;
    if (t >= total) return;
    const unsigned n = t / NR_LEVELS;          // point  (24 lanes share one n)
    const unsigned l = t - n * NR_LEVELS;      // level  (fastest varying)

    // --- lift to hyperplane ------------------------------------------------
    const float p0 = positions[n * 3 + 0];
    const float p1 = positions[n * 3 + 1];
    const float p2 = positions[n * 3 + 2];

    const float cf0 = (p0 + s_shift[l * 3 + 0]) * s_sf[l * 3 + 0];
    const float cf1 = (p1 + s_shift[l * 3 + 1]) * s_sf[l * 3 + 1];
    const float cf2 = (p2 + s_shift[l * 3 + 2]) * s_sf[l * 3 + 2];

    float e[DP1];
    const float s21 = cf2 + cf1;               // reversed-cumsum association
    e[0] = s21 + cf0;
    e[1] = s21 - cf0;
    e[2] = cf2 - 2.0f * cf1;
    e[3] = -3.0f * cf2;

    // --- nearest remainder-0 point ----------------------------------------
    int   rem0[DP1];
    float diff[DP1];
    int   ssum = 0;
#pragma unroll
    for (int i = 0; i < DP1; ++i) {
        float v   = e[i] * 0.25f;
        float up  = ceilf(v)  * 4.0f;
        float dn  = floorf(v) * 4.0f;
        float r0f = (up - e[i] < e[i] - dn) ? up : dn;
        int   r0  = (int)r0f;
        rem0[i]   = r0;
        ssum     += r0;
        diff[i]   = e[i] - r0f;                // pre-fixup diff, as reference
    }
    ssum /= DP1;  // C trunc-toward-zero == sign(s)*(|s|//4)

    // --- ranks + fix-up ----------------------------------------------------
    int rank[DP1];
#pragma unroll
    for (int i = 0; i < DP1; ++i) {
        int rk = 0;
#pragma unroll
        for (int k = 0; k < DP1; ++k)
            rk += (diff[k] > diff[i]) || ((diff[k] == diff[i]) && (k < i));
        rk += ssum;
        const int adj = ((rk < 0) ? 4 : 0) - ((rk > POS_DIM) ? 4 : 0);
        rank[i]  = rk + adj;
        rem0[i] += adj;
    }

    // --- hash the 4 simplex corners EARLY, prefetch the gather lines -------
    const float* Lbase = lattice + (size_t)l * (CAPACITY * NR_FEAT);
    unsigned off[DP1];
#pragma unroll
    for (int r = 0; r < DP1; ++r) {
        const int k0 = rem0[0] + r - 4 * (rank[0] > POS_DIM - r);
        const int k1 = rem0[1] + r - 4 * (rank[1] > POS_DIM - r);
        const int k2 = rem0[2] + r - 4 * (rank[2] > POS_DIM - r);
        const unsigned h = (unsigned)k0
                         ^ ((unsigned)k1 * 2654435761u)
                         ^ ((unsigned)k2 * 805459861u);
        off[r] = (h & (CAPACITY - 1u)) * NR_FEAT;
        // gfx1250 global_prefetch_b8: warm L0/L2 while we compute the weights
        __builtin_prefetch(Lbase + off[r], 0, 3);
    }

    // --- barycentric weights ------------------------------------------------
    // oh = one_hot(3-rank) - one_hot(4-rank); since b = a+1, (b==k) == (a==k-1)
    // -> running-select form: per i only 5 selects + 5 subs + 5 adds.
    // rank is provably in [0,3] after fix-up (elevated coords are zero-sum),
    // so this matches the reference one_hot semantics exactly.
    float bary[DP1 + 1] = {0.f, 0.f, 0.f, 0.f, 0.f};
#pragma unroll
    for (int i = 0; i < DP1; ++i) {
        const float delta = (e[i] - (float)rem0[i]) * 0.25f;
        const int a = POS_DIM - rank[i];       // 3 - rank, in [0,3]
        float prev = 0.f;                      // (a == -1) never true
#pragma unroll
        for (int k = 0; k <= DP1; ++k) {
            const float cur = (a == k) ? delta : 0.f;
            bary[k] += cur - prev;
            prev = cur;
        }
    }
    bary[0] += 1.0f + bary[DP1];

    // --- the 4 float2 gathers (L2-resident table) ---------------------------
    const v2f f0 = *(const v2f*)(Lbase + off[0]);
    const v2f f1 = *(const v2f*)(Lbase + off[1]);
    const v2f f2 = *(const v2f*)(Lbase + off[2]);
    const v2f f3 = *(const v2f*)(Lbase + off[3]);

    const float aw = s_anneal[l];
    v2f o;
    o.x = (f0.x * bary[0] + f1.x * bary[1] + f2.x * bary[2] + f3.x * bary[3]) * aw;
    o.y = (f0.y * bary[0] + f1.y * bary[1] + f2.y * bary[2] + f3.y * bary[3]) * aw;

    // Non-temporal (TH_NT) streaming store: don't evict the lattice from L2.
    v2f* dst = (v2f*)(out + (size_t)n * (NR_LEVELS * NR_FEAT) + l * NR_FEAT);
    __builtin_nontemporal_store(o, dst);
}

// ---------------------------------------------------------------------------
extern "C" void kernel_launch(void* const* d_in, const int* in_sizes, int n_in,
                              void* d_out, int out_size, void* d_ws, size_t ws_size,
                              hipStream_t stream)
{
    const float* positions = (const float*)d_in[0];   // [N,3]
    const float* anneal    = (const float*)d_in[1];   // [24]
    const float* lattice   = (const float*)d_in[2];   // [24,262144,2]
    const float* shift     = (const float*)d_in[3];   // [24,3]
    float*       out       = (float*)d_out;
    float*       sf        = (float*)d_ws;            // 24*3 floats scratch

    const unsigned n_points = (unsigned)(in_sizes[0] / POS_DIM);

    hipLaunchKernelGGL(permuto_init_sf, dim3(1), dim3(32), 0, stream, sf);

    const unsigned total = n_points * NR_LEVELS;
    const unsigned block = 256;
    const unsigned grid  = (total + block - 1) / block;
    hipLaunchKernelGGL(permuto_encode, dim3(grid), dim3(block), 0, stream,
                       positions, anneal, lattice, shift, sf, out, n_points);
}